// ClassPerformanceLoss_31370441130518
// MI455X (gfx1250) — compile-verified
//
#include <hip/hip_runtime.h>

typedef float v2f __attribute__((ext_vector_type(2)));
typedef float v8f __attribute__((ext_vector_type(8)));

#define N_CLASSES 128
#define FLT_BIG 3.402823466e+38f

__global__ void cpl_zero_ws(float* ws) {
    int i = blockIdx.x * blockDim.x + threadIdx.x;
    if (i < 2 * N_CLASSES) ws[i] = 0.0f;
}

__device__ __forceinline__ float sel8(v8f v, int k) {
    float r = v[0];
    r = (k == 1) ? v[1] : r;
    r = (k == 2) ? v[2] : r;
    r = (k == 3) ? v[3] : r;
    r = (k == 4) ? v[4] : r;
    r = (k == 5) ? v[5] : r;
    r = (k == 6) ? v[6] : r;
    r = (k == 7) ? v[7] : r;
    return r;
}

__global__ __launch_bounds__(1024)
void cpl_main(const float* __restrict__ yhat, const float* __restrict__ y,
              float* __restrict__ ws_sum, float* __restrict__ ws_cnt, int B) {
    __shared__ float s_sum[N_CLASSES];
    __shared__ float s_cnt[N_CLASSES];
    for (int i = threadIdx.x; i < N_CLASSES; i += blockDim.x) { s_sum[i] = 0.0f; s_cnt[i] = 0.0f; }
    __syncthreads();

    const int lane  = threadIdx.x & 31;
    const int r     = lane & 15;     // row within tile handled by lane pair (r, r+16)
    const int kh    = lane >> 4;     // which 64-class half this lane streams
    const int gwave = (blockIdx.x * blockDim.x + threadIdx.x) >> 5;
    const int nwave = (gridDim.x * blockDim.x) >> 5;
    const int ntile = (B + 15) >> 4;

    const v2f ones = {1.0f, 1.0f};   // B matrix (4x16) of all ones -> D cols = row sums

    for (int tile = gwave; tile < ntile; tile += nwave) {
        const int row  = tile * 16 + r;
        const int vrow = (row < B) ? row : (B - 1);      // clamp: keep EXEC all-1s for WMMA
        const float4* ph = (const float4*)(yhat + (size_t)vrow * N_CLASSES + kh * 64);
        const float4* py = (const float4*)(y    + (size_t)vrow * N_CLASSES + kh * 64);

        float m = -FLT_BIG, s = 0.0f;                    // online logsumexp state
        float bestV = -FLT_BIG; int bestI = 0;           // argmax(y) state
        v8f accY = {0,0,0,0,0,0,0,0};                    // WMMA chain: rowsum(y)
        v8f accP = {0,0,0,0,0,0,0,0};                    // WMMA chain: rowsum(y*yhat)

        #pragma unroll
        for (int t = 0; t < 16; ++t) {
            const float4 hv = ph[t];                     // global_load_b128, streams 16B/lane
            const float4 yv = py[t];

            // --- VALU: numerically stable online logsumexp over yhat ---
            float m4 = fmaxf(fmaxf(hv.x, hv.y), fmaxf(hv.z, hv.w));
            float mn = fmaxf(m, m4);
            s = s * __expf(m - mn)
              + __expf(hv.x - mn) + __expf(hv.y - mn)
              + __expf(hv.z - mn) + __expf(hv.w - mn);
            m = mn;

            // --- VALU: argmax(y), first-occurrence via strict > on increasing idx ---
            int c0 = kh * 64 + t * 4;
            if (yv.x > bestV) { bestV = yv.x; bestI = c0;     }
            if (yv.y > bestV) { bestV = yv.y; bestI = c0 + 1; }
            if (yv.z > bestV) { bestV = yv.z; bestI = c0 + 2; }
            if (yv.w > bestV) { bestV = yv.w; bestI = c0 + 3; }

            // --- matrix pipe: exact-f32 MAC reductions (co-executes with VALU) ---
            v2f aY0 = { yv.x,        yv.y        };
            v2f aP0 = { yv.x * hv.x, yv.y * hv.y };
            accY = __builtin_amdgcn_wmma_f32_16x16x4_f32(false, aY0, false, ones, (short)0, accY, false, false);
            accP = __builtin_amdgcn_wmma_f32_16x16x4_f32(false, aP0, false, ones, (short)0, accP, false, false);
            v2f aY1 = { yv.z,        yv.w        };
            v2f aP1 = { yv.z * hv.z, yv.w * hv.w };
            accY = __builtin_amdgcn_wmma_f32_16x16x4_f32(false, aY1, false, ones, (short)0, accY, false, false);
            accP = __builtin_amdgcn_wmma_f32_16x16x4_f32(false, aP1, false, ones, (short)0, accP, false, false);
        }

        // --- combine the two 64-class halves (lane <-> lane^16), wave32 ---
        float pm = __shfl_xor(m, 16);
        float ps = __shfl_xor(s, 16);
        float mn = fmaxf(m, pm);
        float st = s * __expf(m - mn) + ps * __expf(pm - mn);
        float lse = mn + __logf(st);

        float pbV = __shfl_xor(bestV, 16);
        int   pbI = __shfl_xor(bestI, 16);
        if (pbV > bestV) { bestV = pbV; bestI = pbI; }   // tie keeps lower idx (local half)

        // --- extract row sums from D layout: lanes 0-15 hold rows 0-7 (VGPR m),
        //     lanes 16-31 hold rows 8-15; all D columns equal since B == ones ---
        const int k = lane & 7;
        float pY = sel8(accY, k);
        float pP = sel8(accP, k);
        float oY = __shfl_xor(pY, 16);
        float oP = __shfl_xor(pP, 16);
        float sumY = (r < 8) ? pY : oY;
        float dotP = (r < 8) ? pP : oP;

        // --- per-row scatter into LDS bins (lanes 0-15 own rows) ---
        if (lane < 16 && row < B) {
            float loss = lse * sumY - dotP;              // -sum(y*(yhat-lse))
            atomicAdd(&s_sum[bestI], loss);              // ds_add_f32
            atomicAdd(&s_cnt[bestI], 1.0f);
        }
    }

    __syncthreads();
    for (int i = threadIdx.x; i < N_CLASSES; i += blockDim.x) {
        atomicAdd(&ws_sum[i], s_sum[i]);                 // global_atomic_add_f32
        atomicAdd(&ws_cnt[i], s_cnt[i]);
    }
}

__global__ void cpl_finalize(const float* __restrict__ ws_sum,
                             const float* __restrict__ ws_cnt,
                             float* __restrict__ out) {
    int c = threadIdx.x;
    if (c < N_CLASSES) {
        float cnt = ws_cnt[c];
        out[c] = (cnt > 0.0f) ? (ws_sum[c] / cnt) : 0.0f;
    }
}

extern "C" void kernel_launch(void* const* d_in, const int* in_sizes, int n_in,
                              void* d_out, int out_size, void* d_ws, size_t ws_size,
                              hipStream_t stream) {
    const float* yhat = (const float*)d_in[0];
    const float* y    = (const float*)d_in[1];
    float* out    = (float*)d_out;
    float* ws_sum = (float*)d_ws;
    float* ws_cnt = ws_sum + N_CLASSES;

    const int B = in_sizes[0] / N_CLASSES;
    const int ntile = (B + 15) / 16;

    cpl_zero_ws<<<1, 256, 0, stream>>>(ws_sum);

    int blocks = (ntile + 31) / 32;    // 32 waves per 1024-thread block
    if (blocks > 256) blocks = 256;    // cap: amortize LDS-bin flush atomics
    if (blocks < 1) blocks = 1;
    cpl_main<<<blocks, 1024, 0, stream>>>(yhat, y, ws_sum, ws_cnt, B);

    cpl_finalize<<<1, 128, 0, stream>>>(ws_sum, ws_cnt, out);
}